// AdLifNeuron_84155589198565
// MI455X (gfx1250) — compile-verified
//
#include <hip/hip_runtime.h>

// AdLIF neuron scan: x[B,T,D] -> spikes[B,T,D], sequential over T, parallel over B*D.
// Memory-bound streaming kernel: float4 columns, double-buffered register pipeline,
// NT load/store hints, global_prefetch_b8 into L2 ahead of demand loads.

typedef __attribute__((ext_vector_type(4))) float v4f;

#define B_SIZE   32
#define T_STEPS  512
#define D_DIM    1024
#define ROW_F4   (D_DIM / 4)          // 256 float4 per (b,t) row; row stride = 4 KB
#define NCOLS    (B_SIZE * ROW_F4)    // 8192 float4 columns
#define UNROLL   16                   // timesteps per chunk (x2 buffers = 128 VGPRs)
#define NCHUNKS  (T_STEPS / UNROLL)   // 32
#define PF_AHEAD 3                    // prefetch 3 chunks (48 timesteps) ahead

#define ALPHA_MEM 0.9512294245007140f  // exp(-1/20)
#define ALPHA_ADP 0.9950124791926823f  // exp(-1/200)
#define BETA_ADP  0.1f
#define THR       1.0f

__global__ __launch_bounds__(64, 1)
void AdLifNeuron_scan_kernel(const float* __restrict__ xg, float* __restrict__ og) {
    const int col = blockIdx.x * blockDim.x + threadIdx.x;   // 0..NCOLS-1
    const int b   = col >> 8;                                // col / ROW_F4
    const int d4  = col & (ROW_F4 - 1);

    const v4f* __restrict__ xp = (const v4f*)xg + (size_t)b * T_STEPS * ROW_F4 + d4;
    v4f*       __restrict__ op = (v4f*)og       + (size_t)b * T_STEPS * ROW_F4 + d4;

    v4f v = {0.f, 0.f, 0.f, 0.f};   // membrane potential
    v4f a = {0.f, 0.f, 0.f, 0.f};   // adaptive threshold state

    v4f buf[2][UNROLL];

    // Preload chunk 0 (non-temporal: each element is read exactly once).
#pragma unroll
    for (int u = 0; u < UNROLL; ++u)
        buf[0][u] = __builtin_nontemporal_load(xp + (size_t)u * ROW_F4);

    for (int c = 0; c < NCHUNKS; ++c) {
        const int cur = c & 1;
        const int nxt = cur ^ 1;

        // Counter-free, VGPR-free L2 prefetch several chunks ahead
        // (lowers to global_prefetch_b8 on gfx1250).
        if (c + PF_AHEAD < NCHUNKS) {
            const int tp = (c + PF_AHEAD) * UNROLL;
#pragma unroll
            for (int u = 0; u < UNROLL; ++u)
                __builtin_prefetch((const void*)(xp + (size_t)(tp + u) * ROW_F4), 0, 1);
        }

        // Issue next chunk's loads before computing the current chunk so the
        // serial (v,a) dependence chain overlaps the memory latency.
        if (c + 1 < NCHUNKS) {
            const int tn = (c + 1) * UNROLL;
#pragma unroll
            for (int u = 0; u < UNROLL; ++u)
                buf[nxt][u] = __builtin_nontemporal_load(xp + (size_t)(tn + u) * ROW_F4);
        }

        const int t0 = c * UNROLL;
#pragma unroll
        for (int u = 0; u < UNROLL; ++u) {
            const v4f xt = buf[cur][u];
            v = ALPHA_MEM * v + xt;                 // v_fma
            const v4f arg = v - (THR + BETA_ADP * a); // v - cur_thr (v_fma + v_sub)
            v4f s;
#pragma unroll
            for (int k = 0; k < 4; ++k)             // spike_fn: (arg >= THR) ? 1 : 0
                s[k] = (arg[k] >= THR) ? 1.0f : 0.0f;  // v_cmp + v_cndmask
            a = ALPHA_ADP * a + s;                  // v_fma
            v = v - s;                              // reset 'subtract' (s * THR, THR=1)
            __builtin_nontemporal_store(s, op + (size_t)(t0 + u) * ROW_F4);
        }
    }
}

extern "C" void kernel_launch(void* const* d_in, const int* in_sizes, int n_in,
                              void* d_out, int out_size, void* d_ws, size_t ws_size,
                              hipStream_t stream) {
    (void)in_sizes; (void)n_in; (void)d_ws; (void)ws_size; (void)out_size;
    const float* x = (const float*)d_in[0];
    float* out = (float*)d_out;

    // 8192 float4-columns; 64-thread (2-wave) blocks -> 128 workgroups spread
    // across WGPs for aggregate L2/HBM bandwidth.
    const int threads = 64;
    const int blocks  = NCOLS / threads;  // 128
    AdLifNeuron_scan_kernel<<<blocks, threads, 0, stream>>>(x, out);
}